// PeakLocalMax_27874337751651
// MI455X (gfx1250) — compile-verified
//
#include <hip/hip_runtime.h>

// PeakLocalMax for MI455X (gfx1250, wave32).
// 41x41 SAME max-pool == separable 41-window sliding max along W then H.
// Sliding-41 max via log-doubling: win41(x) = max(M32(x), M32(x+9)),
// M32 built by 5 doubling steps. LDS ping-pong per tile; async global->LDS
// staging (CDNA5 ASYNCcnt path) with L2-resident halo re-reads (input 134MB
// fits the 192MB L2). Memory-bound: ~268MB essential traffic -> ~11.5us ideal.

#define IMG_H   1024
#define IMG_W   1024
#define NCH     2
#define HALO    20          // MIN_DISTANCE
#define THRESH  0.5f

#define TX      32
#define TY      64
#define KCOLS   (TX + 2*HALO)   // 72 staged columns
#define RROWS   (TY + 2*HALO)   // 104 staged rows
#define NTHREADS 256

#if defined(__HIP_DEVICE_COMPILE__) && defined(__gfx1250__) && \
    __has_builtin(__builtin_amdgcn_global_load_async_to_lds_b32)
#define USE_ASYNC_COPY 1
#else
#define USE_ASYNC_COPY 0
#endif

typedef __attribute__((address_space(1))) int* gas_i32p;
typedef __attribute__((address_space(3))) int* las_i32p;

__device__ __forceinline__ void async_wait0() {
#if USE_ASYNC_COPY
#if __has_builtin(__builtin_amdgcn_s_wait_asynccnt)
  __builtin_amdgcn_s_wait_asynccnt(0);
#else
  asm volatile("s_wait_asynccnt 0" ::: "memory");
#endif
#endif
}

// dst[r][k] = max(src[r][k], src[r][k + off]) for k < len, all rows
__device__ __forceinline__ void hstep(const float* __restrict__ src,
                                      float* __restrict__ dst,
                                      int len, int off, int tid) {
  for (int i = tid; i < RROWS * len; i += NTHREADS) {
    int r = i / len;
    int k = i - r * len;
    int base = r * KCOLS + k;
    dst[base] = fmaxf(src[base], src[base + off]);
  }
}

// dst[r][x] = max(src[r][x], src[r + off][x]) for r < nrows, x < TX
__device__ __forceinline__ void vstep(const float* __restrict__ src,
                                      float* __restrict__ dst,
                                      int nrows, int off, int tid) {
  for (int i = tid; i < nrows * TX; i += NTHREADS) {
    int r = i / TX;
    int x = i - r * TX;
    dst[r * KCOLS + x] = fmaxf(src[r * KCOLS + x], src[(r + off) * KCOLS + x]);
  }
}

__global__ __launch_bounds__(NTHREADS)
void PeakLocalMax_27874337751651_kernel(const float* __restrict__ in,
                                        float* __restrict__ out) {
  __shared__ float Abuf[RROWS * KCOLS];   // 29952 B
  __shared__ float Bbuf[RROWS * KCOLS];   // 29952 B

  const int tid = threadIdx.x;
  const int x0  = blockIdx.x * TX;
  const int y0  = blockIdx.y * TY;
  const int ch  = blockIdx.z & 1;
  const int b   = blockIdx.z >> 1;
  const size_t planeBase = (size_t)b * IMG_H * IMG_W; // in float2 pixels
  const float NEG_INF = -__builtin_huge_valf();

  // ---- Stage (TY+40) x (TX+40) single-channel region into LDS ----
  for (int i = tid; i < RROWS * KCOLS; i += NTHREADS) {
    int r  = i / KCOLS;
    int k  = i - r * KCOLS;
    int gy = y0 + r - HALO;
    int gx = x0 + k - HALO;
    if ((unsigned)gy < (unsigned)IMG_H && (unsigned)gx < (unsigned)IMG_W) {
      const float* gp = in + ((planeBase + (size_t)gy * IMG_W + gx) * NCH + ch);
#if USE_ASYNC_COPY
      __builtin_amdgcn_global_load_async_to_lds_b32(
          (gas_i32p)const_cast<float*>(gp),
          (las_i32p)&Abuf[i],
          0 /*offset*/, 0 /*cpol*/);
#else
      Abuf[i] = *gp;
#endif
    } else {
      Abuf[i] = NEG_INF;
    }
  }
  async_wait0();
  __syncthreads();

  // ---- Horizontal sliding-41 max via doubling (per staged row) ----
  hstep(Abuf, Bbuf, 71, 1, tid);  __syncthreads();  // window 2
  hstep(Bbuf, Abuf, 69, 2, tid);  __syncthreads();  // window 4
  hstep(Abuf, Bbuf, 65, 4, tid);  __syncthreads();  // window 8
  hstep(Bbuf, Abuf, 57, 8, tid);  __syncthreads();  // window 16
  hstep(Abuf, Bbuf, 41, 16, tid); __syncthreads();  // window 32
  // win41 row max for output col x: max(M32[x], M32[x+9]) -> Abuf[r][x], x<TX
  hstep(Bbuf, Abuf, TX, 9, tid);  __syncthreads();

  // ---- Vertical sliding-41 max via doubling over 104 row-max rows ----
  vstep(Abuf, Bbuf, 103, 1, tid);  __syncthreads(); // window 2
  vstep(Bbuf, Abuf, 101, 2, tid);  __syncthreads(); // window 4
  vstep(Abuf, Bbuf,  97, 4, tid);  __syncthreads(); // window 8
  vstep(Bbuf, Abuf,  89, 8, tid);  __syncthreads(); // window 16
  vstep(Abuf, Bbuf,  73, 16, tid); __syncthreads(); // window 32

  // ---- Final combine + peak test + store ----
  for (int i = tid; i < TY * TX; i += NTHREADS) {
    int y = i / TX;
    int x = i - y * TX;
    float m = fmaxf(Bbuf[y * KCOLS + x], Bbuf[(y + 9) * KCOLS + x]); // window 41
    size_t gi = (planeBase + (size_t)(y0 + y) * IMG_W + (x0 + x)) * NCH + ch;
    float v = in[gi];            // center value (L0/L2 hot)
    out[gi] = (v == m && v > THRESH) ? 1.0f : 0.0f;
  }
}

extern "C" void kernel_launch(void* const* d_in, const int* in_sizes, int n_in,
                              void* d_out, int out_size, void* d_ws, size_t ws_size,
                              hipStream_t stream) {
  (void)n_in; (void)out_size; (void)d_ws; (void)ws_size;
  const float* in = (const float*)d_in[0];
  float* out = (float*)d_out;
  int nb = in_sizes[0] / (IMG_H * IMG_W * NCH);
  if (nb < 1) nb = 1;
  dim3 grid(IMG_W / TX, IMG_H / TY, nb * NCH);
  PeakLocalMax_27874337751651_kernel<<<grid, dim3(NTHREADS), 0, stream>>>(in, out);
}